// LSTM_DOUBLE_ATT_STACK_PARALLEL_DROPOUT_57990648430690
// MI455X (gfx1250) — compile-verified
//
#include <hip/hip_runtime.h>
#include <math.h>
#include <stdint.h>

// ---------------------------------------------------------------------------
// LSTM + double attention for MI455X (gfx1250, wave32, WMMA + TDM).
// fp32 GEMMs on v_wmma_f32_16x16x4_f32; att tiles staged via tensor_load_to_lds.
// ---------------------------------------------------------------------------

typedef __attribute__((ext_vector_type(2))) float v2f;
typedef __attribute__((ext_vector_type(8))) float v8f;
typedef __attribute__((ext_vector_type(4))) unsigned int v4u;
typedef __attribute__((ext_vector_type(8))) unsigned int v8u;

constexpr int Bn   = 128;
constexpr int An   = 512;
constexpr int Rn   = 512;
constexpr int INn  = 512;
constexpr int Ln   = 2;
constexpr int Pn   = 2;
constexpr int OUTn = 10000;
constexpr int G4R  = 4 * Rn;   // 2048

#define WMMA_F32X4(acc, a, b)                                                  \
  acc = __builtin_amdgcn_wmma_f32_16x16x4_f32(false, (a), false, (b),          \
                                              (short)0, (acc), false, false)

__device__ __forceinline__ float sigm_(float x) { return 1.0f / (1.0f + expf(-x)); }

// ---------------------------------------------------------------------------
// Generic C = A(MxK) * W(NxK)^T + bias. Each wave: one 16-row M tile x NT
// 16-col N tiles -> NT independent WMMA accumulator chains per A fragment.
// ---------------------------------------------------------------------------
template <int NT>
__global__ __launch_bounds__(256) void gemm_wmma_nt_kernel(
    const float* __restrict__ A, const float* __restrict__ W,
    const float* __restrict__ bias, float* __restrict__ C,
    int M, int N, int K) {
  const int groupsN = (N >> 4) / NT;
  const int numT    = (M >> 4) * groupsN;
  const int t = blockIdx.x * 8 + (threadIdx.x >> 5);
  if (t >= numT) return;                         // wave-uniform (EXEC safe)
  const int m0 = (t / groupsN) << 4;
  const int n0 = (t % groupsN) * (16 * NT);
  const int lane = threadIdx.x & 31;
  const int half = lane >> 4;
  const int l16  = lane & 15;

  const float* aRow = A + (size_t)(m0 + l16) * K + 2 * half;
  const float* wRow[NT];
  v8f acc[NT];
  const v8f zero = {};
#pragma unroll
  for (int j = 0; j < NT; ++j) {
    wRow[j] = W + (size_t)(n0 + 16 * j + l16) * K + 2 * half;
    acc[j] = zero;
  }
  for (int k = 0; k < K; k += 4) {
    v2f af = *(const v2f*)(aRow + k);
#pragma unroll
    for (int j = 0; j < NT; ++j) {
      v2f bf = *(const v2f*)(wRow[j] + k);
      WMMA_F32X4(acc[j], af, bf);
    }
  }
#pragma unroll
  for (int j = 0; j < NT; ++j) {
    const int n  = n0 + 16 * j + l16;
    const float bn = bias ? bias[n] : 0.0f;
#pragma unroll
    for (int v = 0; v < 8; ++v)
      C[(size_t)(m0 + v + 8 * half) * N + n] = acc[j][v] + bn;
  }
}

// ---------------------------------------------------------------------------
// Fused attention scores: scores[b,a] = bd + sum_k tanh( att[b,a,:].Wa[k,:]
//                                              + ba[k] + hh[b,k] ) * Wd[k]
// One block per (b, 16-row a tile). The 16x512 fp32 att tile is DMA'd into
// LDS by the Tensor Data Mover (wave 0 issues tensor_load_to_lds). TDM pad
// (2 DWORDs per 256) yields an LDS row stride of 516 DWORDs: bank-conflict-
// free A-fragment reads, with a 2-DWORD gap at column 256.
// Each of the 8 waves owns 64 contiguous k columns -> 4 WMMA chains.
// ---------------------------------------------------------------------------
__global__ __launch_bounds__(256) void attn_scores_kernel(
    const float* __restrict__ att, const float* __restrict__ Wa,
    const float* __restrict__ ba, const float* __restrict__ hh,
    const float* __restrict__ Wd, const float* __restrict__ bd,
    float* __restrict__ scores) {
  constexpr int LDP = Rn + 4;                    // 516-DWORD padded row stride
  __shared__ float ldsAtt[16 * LDP];             // 33 KB of 320 KB WGP LDS
  __shared__ float swave[8][16];

  const int b   = blockIdx.x >> 5;               // 32 a-tiles per batch row
  const int a0  = (blockIdx.x & 31) << 4;
  const int tid = threadIdx.x;

  // ---- TDM stage: 16 rows x 512 fp32, global -> LDS ----------------------
  if (tid < 32) {                                // wave 0 only (TDM ignores EXEC)
    const uint64_t gaddr = (uint64_t)(uintptr_t)(att + ((size_t)b * An + a0) * Rn);
    const uint32_t laddr = (uint32_t)(uintptr_t)(&ldsAtt[0]);  // LDS offset
    v4u g0;
    g0[0] = 1u;                                          // count=1, user desc
    g0[1] = laddr;                                       // lds_addr[63:32]
    g0[2] = (uint32_t)gaddr;                             // global_addr lo
    g0[3] = ((uint32_t)(gaddr >> 32) & 0x01FFFFFFu)      // global_addr hi
            | (2u << 30);                                // type=2 (image)
    v8u g1;
    g1[0] = (2u << 16)        // data_size = 4 bytes
          | (1u << 20)        // pad_enable
          | (7u << 22)        // pad_interval: every 256 DWORDs
          | (1u << 25);       // pad_amount: 2 DWORDs
    g1[1] = (uint32_t)Rn << 16;    // tensor_dim0 = 512 (bits 79:48)
    g1[2] = 16u << 16;             // tensor_dim1 = 16  (bits 111:80)
    g1[3] = (uint32_t)Rn << 16;    // tile_dim0 = 512   (bits 127:112)
    g1[4] = 16u;                   // tile_dim1 = 16, tile_dim2 = 0
    g1[5] = (uint32_t)Rn;          // tensor_dim0_stride = 512 (lo 32)
    g1[6] = 0u;                    // stride0 hi, stride1 lo
    g1[7] = 0u;                    // stride1 hi
    asm volatile("tensor_load_to_lds %0, %1" ::"s"(g0), "s"(g1) : "memory");
    __builtin_amdgcn_s_wait_tensorcnt(0);
  }
  __syncthreads();

  const int wave = tid >> 5;
  const int lane = tid & 31;
  const int half = lane >> 4;
  const int l16  = lane & 15;
  const float* hhB = hh + (size_t)b * An;
  const int kbase  = wave << 6;                  // 64 k columns per wave

  const float* w0 = Wa + (size_t)(kbase +  0 + l16) * Rn + 2 * half;
  const float* w1 = Wa + (size_t)(kbase + 16 + l16) * Rn + 2 * half;
  const float* w2 = Wa + (size_t)(kbase + 32 + l16) * Rn + 2 * half;
  const float* w3 = Wa + (size_t)(kbase + 48 + l16) * Rn + 2 * half;
  __builtin_prefetch(w0, 0, 3);
  __builtin_prefetch(w1, 0, 3);
  __builtin_prefetch(w2, 0, 3);
  __builtin_prefetch(w3, 0, 3);
  const float* aBase = &ldsAtt[l16 * LDP + 2 * half];

  const v8f zero = {};
  v8f acc0 = zero, acc1 = zero, acc2 = zero, acc3 = zero;
  for (int r = 0; r < 256; r += 4) {             // first half-row (no gap)
    v2f af = *(const v2f*)(aBase + r);
    WMMA_F32X4(acc0, af, *(const v2f*)(w0 + r));
    WMMA_F32X4(acc1, af, *(const v2f*)(w1 + r));
    WMMA_F32X4(acc2, af, *(const v2f*)(w2 + r));
    WMMA_F32X4(acc3, af, *(const v2f*)(w3 + r));
  }
  for (int r = 256; r < Rn; r += 4) {            // second half: skip TDM pad
    v2f af = *(const v2f*)(aBase + r + 2);
    WMMA_F32X4(acc0, af, *(const v2f*)(w0 + r));
    WMMA_F32X4(acc1, af, *(const v2f*)(w1 + r));
    WMMA_F32X4(acc2, af, *(const v2f*)(w2 + r));
    WMMA_F32X4(acc3, af, *(const v2f*)(w3 + r));
  }

  float partial[8];
#pragma unroll
  for (int v = 0; v < 8; ++v) partial[v] = 0.0f;
  auto accum = [&](const v8f& a, int kt) {
    const int   k  = kbase + kt * 16 + l16;      // this lane's k column
    const float hk = hhB[k] + ba[k];
    const float wd = Wd[k];
#pragma unroll
    for (int v = 0; v < 8; ++v)
      partial[v] += tanhf(a[v] + hk) * wd;       // a[v] = (a0+v+8*half, k)
  };
  accum(acc0, 0); accum(acc1, 1); accum(acc2, 2); accum(acc3, 3);

  // reduce over the 16 lanes of each half (columns k of the tile)
#pragma unroll
  for (int v = 0; v < 8; ++v) {
    float s = partial[v];
    s += __shfl_xor(s, 1, 32);
    s += __shfl_xor(s, 2, 32);
    s += __shfl_xor(s, 4, 32);
    s += __shfl_xor(s, 8, 32);
    partial[v] = s;
  }
  if (l16 == 0) {
#pragma unroll
    for (int v = 0; v < 8; ++v) swave[wave][v + 8 * half] = partial[v];
  }
  __syncthreads();

  if (tid < 16) {
    float s = bd[0];
#pragma unroll
    for (int w = 0; w < 8; ++w) s += swave[w][tid];
    scores[(size_t)b * An + a0 + tid] = s;
  }
}

// ---------------------------------------------------------------------------
// softmax over a (A=512) per batch row; deterministic LDS tree reductions.
// ---------------------------------------------------------------------------
__global__ __launch_bounds__(256) void softmax_kernel(
    const float* __restrict__ scores, float* __restrict__ weight) {
  __shared__ float red[256];
  const int b   = blockIdx.x;
  const int tid = threadIdx.x;
  const float* s = scores + (size_t)b * An;
  float m = -3.402823466e38f;
  for (int i = tid; i < An; i += 256) m = fmaxf(m, s[i]);
  red[tid] = m; __syncthreads();
  for (int st = 128; st > 0; st >>= 1) {
    if (tid < st) red[tid] = fmaxf(red[tid], red[tid + st]);
    __syncthreads();
  }
  const float mx = red[0]; __syncthreads();
  float sum = 0.0f;
  for (int i = tid; i < An; i += 256) {
    const float e = expf(s[i] - mx);
    weight[(size_t)b * An + i] = e;
    sum += e;
  }
  red[tid] = sum; __syncthreads();
  for (int st = 128; st > 0; st >>= 1) {
    if (tid < st) red[tid] += red[tid + st];
    __syncthreads();
  }
  const float inv = 1.0f / red[0];
  for (int i = tid; i < An; i += 256) weight[(size_t)b * An + i] *= inv;
}

// ---------------------------------------------------------------------------
// att_res[b,r] = sum_a att[b,a,r] * weight[b,a]  (bandwidth pass, L2-hot att)
// ---------------------------------------------------------------------------
__global__ __launch_bounds__(256) void attres_kernel(
    const float* __restrict__ att, const float* __restrict__ weight,
    float* __restrict__ outv) {
  __shared__ float w[An];
  const int b = blockIdx.x >> 1;                 // 2 r-chunks of 256
  const int r = ((blockIdx.x & 1) << 8) + threadIdx.x;
  for (int i = threadIdx.x; i < An; i += 256) w[i] = weight[(size_t)b * An + i];
  __syncthreads();
  const float* base = att + (size_t)b * An * Rn + r;
  float acc = 0.0f;
  for (int a = 0; a < An; ++a) acc += base[(size_t)a * Rn] * w[a];
  outv[(size_t)b * Rn + r] = acc;
}

// ---------------------------------------------------------------------------
// Gates GEMM: Sums(B x P*4R) = X*Wi^T + H*Wh^T + Ares*Wa2^T + (bi+bh+ba2)
// NT=4 register blocking, three K=512 operand-pair segments per accumulator.
// ---------------------------------------------------------------------------
__global__ __launch_bounds__(256) void gates_gemm_kernel(
    const float* __restrict__ X, const float* __restrict__ H,
    const float* __restrict__ Ares, const float* __restrict__ Wi,
    const float* __restrict__ Wh, const float* __restrict__ Wa2,
    const float* __restrict__ bi, const float* __restrict__ bh,
    const float* __restrict__ ba2, float* __restrict__ Sums) {
  constexpr int N  = Pn * G4R;                   // 4096
  constexpr int NT = 4;
  const int groupsN = (N >> 4) / NT;             // 64
  const int t  = blockIdx.x * 8 + (threadIdx.x >> 5);
  const int m0 = (t / groupsN) << 4;
  const int n0 = (t % groupsN) * (16 * NT);
  const int lane = threadIdx.x & 31;
  const int half = lane >> 4;
  const int l16  = lane & 15;

  const float* aM[3] = {X, H, Ares};
  const float* wM[3] = {Wi, Wh, Wa2};
  const v8f zero = {};
  v8f acc[NT] = {zero, zero, zero, zero};
#pragma unroll
  for (int s = 0; s < 3; ++s) {
    const float* aRow = aM[s] + (size_t)(m0 + l16) * Rn + 2 * half;
    const float* wRow[NT];
#pragma unroll
    for (int j = 0; j < NT; ++j)
      wRow[j] = wM[s] + (size_t)(n0 + 16 * j + l16) * Rn + 2 * half;
    for (int k = 0; k < Rn; k += 4) {
      v2f af = *(const v2f*)(aRow + k);
#pragma unroll
      for (int j = 0; j < NT; ++j) {
        v2f bf = *(const v2f*)(wRow[j] + k);
        WMMA_F32X4(acc[j], af, bf);
      }
    }
  }
#pragma unroll
  for (int j = 0; j < NT; ++j) {
    const int n  = n0 + 16 * j + l16;
    const float bn = bi[n] + bh[n] + ba2[n];
#pragma unroll
    for (int v = 0; v < 8; ++v)
      Sums[(size_t)(m0 + v + 8 * half) * N + n] = acc[j][v] + bn;
  }
}

// ---------------------------------------------------------------------------
// LSTM cell elementwise + mean over P parallel paths.
// ---------------------------------------------------------------------------
__global__ __launch_bounds__(256) void lstm_cell_kernel(
    const float* __restrict__ sums, const float* __restrict__ prev_c,
    float* __restrict__ next_c_out, float* __restrict__ next_h_out) {
  const int idx = blockIdx.x * 256 + threadIdx.x;   // b*R + r
  const int b = idx >> 9;
  const int r = idx & (Rn - 1);
  float csum = 0.0f, hsum = 0.0f;
#pragma unroll
  for (int p = 0; p < Pn; ++p) {
    const float* sp = sums + ((size_t)b * Pn + p) * G4R;
    const float ig = sigm_(sp[r]);
    const float fg = sigm_(sp[Rn + r]);
    const float og = sigm_(sp[2 * Rn + r]);
    const float it = tanhf(sp[3 * Rn + r]);
    const float c  = fg * prev_c[idx] + ig * it;
    csum += c;
    hsum += og * tanhf(c);
  }
  next_c_out[idx] = csum * (1.0f / Pn);
  next_h_out[idx] = hsum * (1.0f / Pn);
}

// ---------------------------------------------------------------------------
// top_h = att_res + next_h; optionally xt_next = x + top_h (layer 0 only).
// ---------------------------------------------------------------------------
__global__ __launch_bounds__(256) void add_top_kernel(
    const float* __restrict__ a, const float* __restrict__ b,
    float* __restrict__ outv, const float* __restrict__ x,
    float* __restrict__ xt) {
  const int idx = blockIdx.x * 256 + threadIdx.x;
  const float t = a[idx] + b[idx];
  outv[idx] = t;
  if (xt) xt[idx] = x[idx] + t;
}

// ---------------------------------------------------------------------------
// log_softmax over OUT=10000 per batch row.
// ---------------------------------------------------------------------------
__global__ __launch_bounds__(256) void logsoftmax_kernel(
    const float* __restrict__ logits, float* __restrict__ outv) {
  __shared__ float red[256];
  const int b   = blockIdx.x;
  const int tid = threadIdx.x;
  const float* z = logits + (size_t)b * OUTn;
  float m = -3.402823466e38f;
  for (int i = tid; i < OUTn; i += 256) m = fmaxf(m, z[i]);
  red[tid] = m; __syncthreads();
  for (int st = 128; st > 0; st >>= 1) {
    if (tid < st) red[tid] = fmaxf(red[tid], red[tid + st]);
    __syncthreads();
  }
  const float mx = red[0]; __syncthreads();
  float sum = 0.0f;
  for (int i = tid; i < OUTn; i += 256) sum += expf(z[i] - mx);
  red[tid] = sum; __syncthreads();
  for (int st = 128; st > 0; st >>= 1) {
    if (tid < st) red[tid] += red[tid + st];
    __syncthreads();
  }
  const float lse = logf(red[0]) + mx;
  for (int i = tid; i < OUTn; i += 256)
    outv[(size_t)b * OUTn + i] = z[i] - lse;
}

// ---------------------------------------------------------------------------
extern "C" void kernel_launch(void* const* d_in, const int* in_sizes, int n_in,
                              void* d_out, int out_size, void* d_ws, size_t ws_size,
                              hipStream_t stream) {
  (void)in_sizes; (void)n_in; (void)out_size; (void)ws_size;
  const float* x      = (const float*)d_in[0];
  const float* att    = (const float*)d_in[1];
  const float* inputs = (const float*)d_in[2];
  const float* a2a_w  = (const float*)d_in[3];
  const float* a2a_b  = (const float*)d_in[4];
  const float* h2a_w  = (const float*)d_in[5];
  const float* h2a_b  = (const float*)d_in[6];
  const float* d2d_w  = (const float*)d_in[7];
  const float* d2d_b  = (const float*)d_in[8];
  const float* a2a1_w = (const float*)d_in[9];
  const float* a2a1_b = (const float*)d_in[10];
  const float* h2a1_w = (const float*)d_in[11];
  const float* h2a1_b = (const float*)d_in[12];
  const float* d2d1_w = (const float*)d_in[13];
  const float* d2d1_b = (const float*)d_in[14];
  const float* i2h_w  = (const float*)d_in[15];
  const float* i2h_b  = (const float*)d_in[16];
  const float* h2h_w  = (const float*)d_in[17];
  const float* h2h_b  = (const float*)d_in[18];
  const float* a2h_w  = (const float*)d_in[19];
  const float* a2h_b  = (const float*)d_in[20];
  const float* proj_w = (const float*)d_in[21];
  const float* proj_b = (const float*)d_in[22];
  float* out = (float*)d_out;

  float* ws     = (float*)d_ws;
  float* hh     = ws;                       // B*A
  float* scores = hh     + Bn * An;         // B*A
  float* weight = scores + Bn * An;         // B*A
  float* attres = weight + Bn * An;         // B*R
  float* next_h = attres + Bn * Rn;         // B*R
  float* xtbuf  = next_h + Bn * Rn;         // B*IN
  float* sums   = xtbuf  + Bn * INn;        // B*P*4R
  float* logits = sums   + (size_t)Bn * Pn * G4R;  // B*OUT

  auto attend = [&](const float* h, const float* Wa, const float* ba,
                    const float* Wh, const float* bh,
                    const float* Wd, const float* bd) {
    // hh = h * Wh^T + bh : M=128, N=512, K=512 -> 64 waves (NT=4) -> 8 blocks
    gemm_wmma_nt_kernel<4><<<8, 256, 0, stream>>>(h, Wh, bh, hh, Bn, An, Rn);
    attn_scores_kernel<<<Bn * (An / 16), 256, 0, stream>>>(att, Wa, ba, hh, Wd,
                                                           bd, scores);
    softmax_kernel<<<Bn, 256, 0, stream>>>(scores, weight);
    attres_kernel<<<Bn * (Rn / 256), 256, 0, stream>>>(att, weight, attres);
  };

  for (int i = 0; i < Ln; ++i) {
    const float* xt     = (i == 0) ? x : xtbuf;
    const float* prev_c = inputs + (size_t)(2 * i) * Bn * Rn;
    const float* prev_h = inputs + (size_t)(2 * i + 1) * Bn * Rn;

    attend(prev_h, a2a_w, a2a_b, h2a_w, h2a_b, d2d_w, d2d_b);

    gates_gemm_kernel<<<64, 256, 0, stream>>>(
        xt, prev_h, attres,
        i2h_w + (size_t)i * Pn * G4R * INn,
        h2h_w + (size_t)i * Pn * G4R * Rn,
        a2h_w + (size_t)i * Pn * G4R * Rn,
        i2h_b + (size_t)i * Pn * G4R,
        h2h_b + (size_t)i * Pn * G4R,
        a2h_b + (size_t)i * Pn * G4R, sums);

    lstm_cell_kernel<<<(Bn * Rn) / 256, 256, 0, stream>>>(
        sums, prev_c, out + (size_t)(2 * i) * Bn * Rn, next_h);

    attend(next_h, a2a1_w, a2a1_b, h2a1_w, h2a1_b, d2d1_w, d2d1_b);

    add_top_kernel<<<(Bn * Rn) / 256, 256, 0, stream>>>(
        attres, next_h, out + (size_t)(2 * i + 1) * Bn * Rn,
        (i == 0) ? x : nullptr, (i == 0) ? xtbuf : nullptr);
  }

  // projection: M=128, N=10000 (=625*16), K=512 -> NT=5 -> 1000 waves
  gemm_wmma_nt_kernel<5><<<125, 256, 0, stream>>>(out + (size_t)3 * Bn * Rn,
                                                  proj_w, proj_b, logits, Bn,
                                                  OUTn, Rn);
  logsoftmax_kernel<<<Bn, 256, 0, stream>>>(logits, out + (size_t)4 * Bn * Rn);
}